// CANConv_19550691131445
// MI455X (gfx1250) — compile-verified
//
#include <hip/hip_runtime.h>
#include <hip/hip_bf16.h>

// ---------------- problem constants ----------------
#define B_    4
#define CIN   32
#define COUT  32
#define Hh    64
#define Ww    64
#define Kc    32
#define Mm    16
#define AREA  9
#define Ss    (Hh * Ww)      // 4096
#define Ff    (CIN * AREA)   // 288
#define KMIT  10

typedef float v2f __attribute__((ext_vector_type(2)));
typedef float v8f __attribute__((ext_vector_type(8)));

// ---------------------------------------------------------------------------
// Kernel 1: feat[b,s,c] = mean over 3x3 zero-padded neighborhood of x[b,c,:,:]
// Layout: feat stored [B][S][CIN] for k-means point access.
// Thread id = (b*CIN + c)*S + s  -> coalesced x reads over s.
// ---------------------------------------------------------------------------
__global__ void feat_kernel(const float* __restrict__ x, float* __restrict__ feat) {
    int id = blockIdx.x * blockDim.x + threadIdx.x;
    int s  = id & (Ss - 1);
    int bc = id >> 12;            // / Ss
    int c  = bc & (CIN - 1);
    int b  = bc >> 5;
    int h = s >> 6, w = s & 63;
    const float* xp = x + ((size_t)(b * CIN + c) * Hh) * Ww;
    float sum = 0.f;
#pragma unroll
    for (int dh = -1; dh <= 1; ++dh)
#pragma unroll
        for (int dw = -1; dw <= 1; ++dw) {
            int hh = h + dh, ww = w + dw;
            if (hh >= 0 && hh < Hh && ww >= 0 && ww < Ww)
                sum += xp[hh * Ww + ww];
        }
    feat[(size_t)(b * Ss + s) * CIN + c] = sum / 9.0f;
}

// ---------------------------------------------------------------------------
// Kernel 2: k-means per batch (1 workgroup of 256 threads per batch).
// Deterministic: assignment in parallel, centroid update via fixed-order scan
// where thread t exclusively owns cluster k=t>>3, channels 4*(t&7)..+3.
// Matches reference: strided linspace init, 10 Lloyd iters, final assign,
// argmin keeps first minimum, empty cluster keeps old centroid.
// ---------------------------------------------------------------------------
__global__ void kmeans_kernel(const float* __restrict__ feat,
                              float* __restrict__ centOut,
                              int* __restrict__ idxOut) {
    __shared__ float cents[Kc][CIN];
    __shared__ unsigned char sidx[Ss];
    const int b   = blockIdx.x;
    const int tid = threadIdx.x;   // 256
    const float* fb = feat + (size_t)b * Ss * CIN;

    // strided init: linspace(0, S-1, K) truncated -> k*(S-1)/(K-1)
    for (int i = tid; i < Kc * CIN; i += 256) {
        int k = i >> 5, c = i & 31;
        int sinit = (k * (Ss - 1)) / (Kc - 1);
        cents[k][c] = fb[(size_t)sinit * CIN + c];
    }
    __syncthreads();

    for (int it = 0; it <= KMIT; ++it) {
        // ---- assignment ----
        for (int s = tid; s < Ss; s += 256) {
            const float* p = fb + (size_t)s * CIN;
            float best = 3.402823e38f;
            int bk = 0;
            for (int k = 0; k < Kc; ++k) {
                float d = 0.f;
#pragma unroll
                for (int c = 0; c < CIN; ++c) {
                    float t = p[c] - cents[k][c];
                    d += t * t;
                }
                if (d < best) { best = d; bk = k; }
            }
            sidx[s] = (unsigned char)bk;
            if (it == KMIT) idxOut[b * Ss + s] = bk;
        }
        __syncthreads();
        if (it == KMIT) break;

        // ---- deterministic centroid update ----
        const int k  = tid >> 3;
        const int c0 = (tid & 7) * 4;
        float a0 = 0.f, a1 = 0.f, a2 = 0.f, a3 = 0.f;
        int cnt = 0;
        for (int s = 0; s < Ss; ++s) {
            if (sidx[s] == (unsigned char)k) {
                const float* p = fb + (size_t)s * CIN + c0;
                a0 += p[0]; a1 += p[1]; a2 += p[2]; a3 += p[3];
                ++cnt;
            }
        }
        if (cnt > 0) {
            float inv = 1.0f / (float)cnt;
            cents[k][c0 + 0] = a0 * inv;
            cents[k][c0 + 1] = a1 * inv;
            cents[k][c0 + 2] = a2 * inv;
            cents[k][c0 + 3] = a3 * inv;
        }
        __syncthreads();
    }

    // export final centroids
    for (int i = tid; i < Kc * CIN; i += 256) {
        int k = i >> 5, c = i & 31;
        centOut[(b * Kc + k) * CIN + c] = cents[k][c];
    }
}

// ---------------------------------------------------------------------------
// Kernel 3: per-cluster heads. One thread per (b,k); 128 threads total.
// kf = relu(relu(cent@W1+b1)@W2+b2); w_* = sigmoid(kf@*_w + *_b)
// bias = relu(relu(cent@bh1)@bh2)@bh3 + b3
// ---------------------------------------------------------------------------
__global__ void head_kernel(const float* __restrict__ cent,
                            const float* __restrict__ kh_w1, const float* __restrict__ kh_b1,
                            const float* __restrict__ kh_w2, const float* __restrict__ kh_b2,
                            const float* __restrict__ area_w, const float* __restrict__ area_b,
                            const float* __restrict__ cin_w, const float* __restrict__ cin_b,
                            const float* __restrict__ cout_w, const float* __restrict__ cout_b,
                            const float* __restrict__ bh_w1, const float* __restrict__ bh_b1,
                            const float* __restrict__ bh_w2, const float* __restrict__ bh_b2,
                            const float* __restrict__ bh_w3, const float* __restrict__ bh_b3,
                            float* __restrict__ wcin, float* __restrict__ warea,
                            float* __restrict__ wcout, float* __restrict__ biasOut) {
    int t = blockIdx.x * blockDim.x + threadIdx.x;
    if (t >= B_ * Kc) return;
    const float* c = cent + t * CIN;

    float kf1[Mm], kf2[Mm];
#pragma unroll
    for (int m = 0; m < Mm; ++m) {
        float a = kh_b1[m];
        for (int i = 0; i < CIN; ++i) a += c[i] * kh_w1[i * Mm + m];
        kf1[m] = fmaxf(a, 0.f);
    }
#pragma unroll
    for (int m = 0; m < Mm; ++m) {
        float a = kh_b2[m];
        for (int i = 0; i < Mm; ++i) a += kf1[i] * kh_w2[i * Mm + m];
        kf2[m] = fmaxf(a, 0.f);
    }
    for (int o = 0; o < CIN; ++o) {
        float a = cin_b[o];
        for (int m = 0; m < Mm; ++m) a += kf2[m] * cin_w[m * CIN + o];
        wcin[t * CIN + o] = 1.0f / (1.0f + expf(-a));
    }
    for (int o = 0; o < AREA; ++o) {
        float a = area_b[o];
        for (int m = 0; m < Mm; ++m) a += kf2[m] * area_w[m * AREA + o];
        warea[t * AREA + o] = 1.0f / (1.0f + expf(-a));
    }
    for (int o = 0; o < COUT; ++o) {
        float a = cout_b[o];
        for (int m = 0; m < Mm; ++m) a += kf2[m] * cout_w[m * COUT + o];
        wcout[t * COUT + o] = 1.0f / (1.0f + expf(-a));
    }
    float h1[Mm], h2[Mm];
#pragma unroll
    for (int m = 0; m < Mm; ++m) {
        float a = bh_b1[m];
        for (int i = 0; i < CIN; ++i) a += c[i] * bh_w1[i * Mm + m];
        h1[m] = fmaxf(a, 0.f);
    }
#pragma unroll
    for (int m = 0; m < Mm; ++m) {
        float a = bh_b2[m];
        for (int i = 0; i < Mm; ++i) a += h1[i] * bh_w2[i * Mm + m];
        h2[m] = fmaxf(a, 0.f);
    }
    for (int o = 0; o < COUT; ++o) {
        float a = bh_b3[o];
        for (int m = 0; m < Mm; ++m) a += h2[m] * bh_w3[m * COUT + o];
        biasOut[t * COUT + o] = a;
    }
}

// ---------------------------------------------------------------------------
// Kernel 4: fused unfold+scale+GEMM+epilogue via V_WMMA_F32_16X16X4_F32.
// Block: 128 threads (4 waves), tile = 32 rows (s values) x 32 cols (cout).
// As[32][289] staged in LDS (stride 289: coprime with 64 banks).
// A frag (16x4 f32, v2f): v0 = K0|K2 across lane halves, v1 = K1|K3.
// B read directly from global base[f*32+c] (36KB, cache-resident).
// C/D (v8f): VGPR r -> rows r (lanes 0-15) / r+8 (lanes 16-31).
// ---------------------------------------------------------------------------
__global__ void __launch_bounds__(128)
dispatch_gemm_kernel(const float* __restrict__ x,
                     const float* __restrict__ base,
                     const int* __restrict__ idx,
                     const float* __restrict__ wcin,
                     const float* __restrict__ warea,
                     const float* __restrict__ wcout,
                     const float* __restrict__ biasIn,
                     float* __restrict__ out) {
    __shared__ float As[32][289];

    const int tid = threadIdx.x;
    const int blk = blockIdx.x;
    const int b   = blk >> 7;           // 128 blocks per batch (4096/32)
    const int s0  = (blk & 127) * 32;

    // stage scaled patch tile: As[row][f] = unfold(x)[s0+row, f] * w_cin * w_area
    for (int i = tid; i < 32 * Ff; i += 128) {
        int row = i / Ff, f = i - row * Ff;
        int s = s0 + row;
        int k = idx[b * Ss + s];
        int cin  = f / AREA;
        int area = f - cin * AREA;
        int h = s >> 6, w = s & 63;
        int hh = h + area / 3 - 1;
        int ww = w + area % 3 - 1;
        float val = 0.f;
        if (hh >= 0 && hh < Hh && ww >= 0 && ww < Ww)
            val = x[((size_t)(b * CIN + cin) * Hh + hh) * Ww + ww];
        As[row][f] = val * wcin[(b * Kc + k) * CIN + cin]
                         * warea[(b * Kc + k) * AREA + area];
    }
    __syncthreads();

    const int wave = tid >> 5;
    const int lane = tid & 31;
    const int lr = lane & 15;
    const int hi = lane >> 4;           // 0 or 1
    const int m0 = (wave >> 1) * 16;    // 0 or 16
    const int n0 = (wave & 1) * 16;     // 0 or 16

    v8f acc = {0.f, 0.f, 0.f, 0.f, 0.f, 0.f, 0.f, 0.f};
    for (int kk = 0; kk < Ff; kk += 4) {
        v2f a, bb;
        a.x  = As[m0 + lr][kk + 2 * hi];
        a.y  = As[m0 + lr][kk + 2 * hi + 1];
        bb.x = base[(kk + 2 * hi) * COUT + n0 + lr];
        bb.y = base[(kk + 2 * hi + 1) * COUT + n0 + lr];
        acc = __builtin_amdgcn_wmma_f32_16x16x4_f32(
            /*neg_a=*/false, a, /*neg_b=*/false, bb,
            /*c_mod=*/(short)0, acc, /*reuse_a=*/false, /*reuse_b=*/false);
    }

    // epilogue: out[b,c,s] = w_cout[b,k,c] * acc + bias[b,k,c]
#pragma unroll
    for (int r = 0; r < 8; ++r) {
        int Mrow = r + 8 * hi;
        int s = s0 + m0 + Mrow;
        int c = n0 + lr;
        int k = idx[b * Ss + s];
        float v = acc[r] * wcout[(b * Kc + k) * COUT + c]
                + biasIn[(b * Kc + k) * COUT + c];
        out[((size_t)(b * COUT + c)) * Ss + s] = v;
    }
}

// ---------------------------------------------------------------------------
extern "C" void kernel_launch(void* const* d_in, const int* in_sizes, int n_in,
                              void* d_out, int out_size, void* d_ws, size_t ws_size,
                              hipStream_t stream) {
    (void)in_sizes; (void)n_in; (void)out_size; (void)ws_size;

    const float* x       = (const float*)d_in[0];
    const float* kh_w1   = (const float*)d_in[1];
    const float* kh_b1   = (const float*)d_in[2];
    const float* kh_w2   = (const float*)d_in[3];
    const float* kh_b2   = (const float*)d_in[4];
    const float* area_w  = (const float*)d_in[5];
    const float* area_b  = (const float*)d_in[6];
    const float* cin_w   = (const float*)d_in[7];
    const float* cin_b   = (const float*)d_in[8];
    const float* cout_w  = (const float*)d_in[9];
    const float* cout_b  = (const float*)d_in[10];
    const float* base    = (const float*)d_in[11];
    const float* bh_w1   = (const float*)d_in[12];
    const float* bh_b1   = (const float*)d_in[13];
    const float* bh_w2   = (const float*)d_in[14];
    const float* bh_b2   = (const float*)d_in[15];
    const float* bh_w3   = (const float*)d_in[16];
    const float* bh_b3   = (const float*)d_in[17];

    // workspace layout (floats)
    float* ws    = (float*)d_ws;
    float* feat  = ws;                                   // B*S*CIN   = 524288
    float* cent  = feat  + (size_t)B_ * Ss * CIN;        // B*K*CIN   = 4096
    float* wcin  = cent  + B_ * Kc * CIN;                // B*K*CIN   = 4096
    float* warea = wcin  + B_ * Kc * CIN;                // B*K*AREA  = 1152
    float* wcout = warea + B_ * Kc * AREA;               // B*K*COUT  = 4096
    float* bias  = wcout + B_ * Kc * COUT;               // B*K*COUT  = 4096
    int*   idx   = (int*)(bias + B_ * Kc * COUT);        // B*S       = 16384

    feat_kernel<<<(B_ * CIN * Ss) / 256, 256, 0, stream>>>(x, feat);
    kmeans_kernel<<<B_, 256, 0, stream>>>(feat, cent, idx);
    head_kernel<<<1, 128, 0, stream>>>(cent,
        kh_w1, kh_b1, kh_w2, kh_b2, area_w, area_b, cin_w, cin_b, cout_w, cout_b,
        bh_w1, bh_b1, bh_w2, bh_b2, bh_w3, bh_b3,
        wcin, warea, wcout, bias);
    dispatch_gemm_kernel<<<(B_ * Ss) / 32, 128, 0, stream>>>(
        x, base, idx, wcin, warea, wcout, bias, (float*)d_out);
}